// SigKernelRidgeRegression_35373350650572
// MI455X (gfx1250) — compile-verified
//
#include <hip/hip_runtime.h>
#include <hip/hip_bf16.h>

typedef float v2f __attribute__((ext_vector_type(2)));
typedef float v8f __attribute__((ext_vector_type(8)));

#define NSER 80          // number of series
#define TLEN 50          // raw series length
#define LREF 197         // refined length: (50-1)*4+1
#define LM1  196         // LREF-1 (PDE grid is LM1 x LM1)
#define NBLK 13          // ceil(197/16) 16-row G blocks; 13*16 = 208
#define NCOLS 208        // padded column count (13 * 16)
#define PADLEN 208       // padded staged-path length (covers all WMMA indices)
#define CPL 7            // columns per lane (32*7 = 224 >= 196)

// -0.5 * log2(e): exp(-0.5*x) == exp2(NEG_HALF_LOG2E * x); v_exp_f32 is exp2.
#define NEG_HALF_LOG2E (-0.7213475204444817f)

// ---------------------------------------------------------------------------
// Kernel 1: refine paths. t = l/196 (exact, since 49/196 = 0.25 per step),
// v = lerp of X, s = t^2 + v^2.
// ---------------------------------------------------------------------------
__global__ void prep_kernel(const float* __restrict__ X,
                            float* __restrict__ T, float* __restrict__ V,
                            float* __restrict__ S) {
    int idx = blockIdx.x * blockDim.x + threadIdx.x;
    if (idx >= NSER * LREF) return;
    int n = idx / LREF, l = idx - n * LREF;
    float pos = 0.25f * (float)l;                 // l * (T-1)/(L-1) = l/4
    int i0 = l >> 2; if (i0 > TLEN - 2) i0 = TLEN - 2;
    float frac = pos - (float)i0;
    const float* xr = X + n * TLEN;
    float v = xr[i0] * (1.0f - frac) + xr[i0 + 1] * frac;
    float t = pos * (1.0f / 49.0f);               // linspace(0,1,50) interp
    T[idx] = t; V[idx] = v; S[idx] = t * t + v * v;
}

// ---------------------------------------------------------------------------
// Kernel 2: one wave per (n,m) pair.
// G strips via WMMA f32 16x16x4:
//   d2[i][j] = [sqX,1,-2t,-2v]_i . [1,sqY,t,v]_j   (K = 4 exactly)
// A/B operand pairs are prebuilt in LDS (indexed by lane "half"), so each
// WMMA operand fetch is a single unconditional ds_load_b64.
// Then exp2 -> LDS G strip, and a wave-parallel affine-scan PDE sweep.
// ---------------------------------------------------------------------------
__global__ __launch_bounds__(32) void sigdp_kernel(
        const float* __restrict__ T, const float* __restrict__ V,
        const float* __restrict__ S, float* __restrict__ Kout) {
    __shared__ float Gbuf[2][16][NCOLS];          // double-buffered G strips
    __shared__ v2f An[2][PADLEN];                 // A operand pairs, series n
    __shared__ v2f Bm[2][PADLEN];                 // B operand pairs, series m

    const int pair = blockIdx.x;
    const int n = pair / NSER, m = pair - n * NSER;
    const int lane = threadIdx.x;
    const int half = lane >> 4;                   // K-pair select: (0,1) vs (2,3)
    const int sub  = lane & 15;

    // Stage prebuilt WMMA operands (padded: tail repeats index 196).
    for (int l = lane; l < PADLEN; l += 32) {
        int ls = l < LREF ? l : LM1;
        float t_n = T[n * LREF + ls], v_n = V[n * LREF + ls], s_n = S[n * LREF + ls];
        float t_m = T[m * LREF + ls], v_m = V[m * LREF + ls], s_m = S[m * LREF + ls];
        v2f a0; a0.x = s_n;          a0.y = 1.0f;
        v2f a1; a1.x = -2.0f * t_n;  a1.y = -2.0f * v_n;
        v2f b0; b0.x = 1.0f;         b0.y = s_m;
        v2f b1; b1.x = t_m;          b1.y = v_m;
        An[0][l] = a0; An[1][l] = a1;
        Bm[0][l] = b0; Bm[1][l] = b1;
    }
    __syncthreads();

    // Compute 16 G rows [16*blk, 16*blk+16) for all 208 padded columns.
    auto computeBlock = [&](int blk) {
        v2f a = An[half][16 * blk + sub];         // one b64 load, no selects
        float* dst = &Gbuf[blk & 1][0][0];
        const int rbase = 8 * half;               // C/D layout: VGPR r -> M = r + 8*half
        for (int jt = 0; jt < NBLK; ++jt) {
            int col = 16 * jt + sub;
            v2f b = Bm[half][col];                // one b64 load, no selects
            v8f acc = {};
            acc = __builtin_amdgcn_wmma_f32_16x16x4_f32(
                false, a, false, b, (short)0, acc, false, false);
            #pragma unroll
            for (int rr = 0; rr < 8; ++rr) {
                // exp(-d2/2) via single mul + v_exp_f32 (hardware exp2)
                float g = __builtin_amdgcn_exp2f(NEG_HALF_LOG2E * acc[rr]);
                dst[(rbase + rr) * NCOLS + col] = g;
            }
        }
    };

    computeBlock(0);
    __syncthreads();

    const int cbase = CPL * lane;                 // this lane's first column
    float kp[8], Gc[8];
    #pragma unroll
    for (int q = 0; q < 8; ++q) {
        kp[q] = 1.0f;                             // k row 0 = ones
        int col = cbase + q; if (col > NCOLS - 1) col = NCOLS - 1;
        Gc[q] = Gbuf[0][0][col];                  // G row 0
    }

    for (int blk = 0; blk < NBLK; ++blk) {
        if (blk + 1 < NBLK) computeBlock(blk + 1);   // one strip ahead
        __syncthreads();
        int i0 = 16 * blk;
        int iend = i0 + 16; if (iend > LM1) iend = LM1;
        for (int i = i0; i < iend; ++i) {
            int g = i + 1;
            const float* grow = &Gbuf[(g >> 4) & 1][g & 15][0];
            float Gn[8];
            #pragma unroll
            for (int q = 0; q < 8; ++q) {
                int col = cbase + q; if (col > NCOLS - 1) col = NCOLS - 1;
                Gn[q] = grow[col];
            }
            // in-lane affine composition: f_j(v) = a*v + b,  b = a*kp[j+1]-c*kp[j]
            float pa[7], pb[7];
            float cpa = 1.0f, cpb = 0.0f;
            #pragma unroll
            for (int q = 0; q < 7; ++q) {
                float A  = Gn[q + 1] - Gn[q] - Gc[q + 1] + Gc[q];
                float A2 = A * A * (1.0f / 12.0f);
                float aa = 1.0f + 0.5f * A + A2;
                float cc = 1.0f - A2;
                float bb = aa * kp[q + 1] - cc * kp[q];
                bool valid = (cbase + q) < LM1;   // identity map beyond j=195
                aa = valid ? aa : 1.0f;
                bb = valid ? bb : 0.0f;
                cpb = aa * cpb + bb;
                cpa = aa * cpa;
                pa[q] = cpa; pb[q] = cpb;
            }
            // Kogge-Stone inclusive scan of affine composition across 32 lanes
            float Ai = cpa, Bi = cpb;
            #pragma unroll
            for (int d = 1; d < 32; d <<= 1) {
                float A2s = __shfl_up(Ai, d, 32);
                float B2s = __shfl_up(Bi, d, 32);
                bool ok = lane >= d;
                A2s = ok ? A2s : 1.0f;
                B2s = ok ? B2s : 0.0f;
                Bi = Ai * B2s + Bi;               // me ∘ prev
                Ai = Ai * A2s;
            }
            float Ae = __shfl_up(Ai, 1, 32);      // exclusive prefix
            float Be = __shfl_up(Bi, 1, 32);
            if (lane == 0) { Ae = 1.0f; Be = 0.0f; }
            float vin = Ae + Be;                  // prefix applied to k_next[0]=1
            kp[0] = vin;
            #pragma unroll
            for (int q = 0; q < 7; ++q) kp[q + 1] = pa[q] * vin + pb[q];
            #pragma unroll
            for (int q = 0; q < 8; ++q) Gc[q] = Gn[q];
        }
    }
    // k[196][196] lives in lane 196/7 = 28, slot 0
    float Kval = __shfl(kp[0], LM1 / CPL, 32);
    if (lane == 0) Kout[n * NSER + m] = Kval;
}

// ---------------------------------------------------------------------------
// Kernel 3: single-block Gauss-Jordan solve of (K + I) alpha = Y (SPD, no
// pivoting needed), then preds = K @ alpha.
// ---------------------------------------------------------------------------
__global__ __launch_bounds__(256) void solve_kernel(
        const float* __restrict__ Kin, const float* __restrict__ Y,
        float* __restrict__ preds, float* __restrict__ alphas) {
    __shared__ float M[NSER][NSER + 1];
    __shared__ float fcol[NSER];
    __shared__ float alph[NSER];
    int t = threadIdx.x;
    for (int idx = t; idx < NSER * NSER; idx += 256) {
        int r = idx / NSER, c = idx - r * NSER;
        M[r][c] = Kin[idx] + (r == c ? 1.0f : 0.0f);   // REG = 1
    }
    for (int r = t; r < NSER; r += 256) M[r][NSER] = Y[r];
    __syncthreads();
    for (int k = 0; k < NSER; ++k) {
        if (t < NSER) fcol[t] = M[t][k] / M[k][k];
        __syncthreads();
        int tx = t & 31, ty = t >> 5;             // 32 cols x 8 row groups
        for (int r = ty; r < NSER; r += 8) {
            if (r == k) continue;                 // pivot row untouched
            float f = fcol[r];
            for (int j = tx; j <= NSER; j += 32)
                M[r][j] -= f * M[k][j];
        }
        __syncthreads();
    }
    if (t < NSER) {
        float al = M[t][NSER] / M[t][t];
        alph[t] = al;
        alphas[t] = al;
    }
    __syncthreads();
    if (t < NSER) {
        float acc = 0.0f;
        for (int j = 0; j < NSER; ++j) acc += Kin[t * NSER + j] * alph[j];
        preds[t] = acc;
    }
}

// ---------------------------------------------------------------------------
extern "C" void kernel_launch(void* const* d_in, const int* in_sizes, int n_in,
                              void* d_out, int out_size, void* d_ws, size_t ws_size,
                              hipStream_t stream) {
    const float* X = (const float*)d_in[0];       // (80, 50)
    const float* Y = (const float*)d_in[1];       // (80,)
    float* out    = (float*)d_out;
    float* preds  = out;                          // 80
    float* Kmat   = out + NSER;                   // 80*80
    float* alphas = out + NSER + NSER * NSER;     // 80

    float* ws = (float*)d_ws;                     // 3 * 80 * 197 floats
    float* T = ws;
    float* V = ws + NSER * LREF;
    float* S = ws + 2 * NSER * LREF;

    int tot = NSER * LREF;
    prep_kernel<<<(tot + 255) / 256, 256, 0, stream>>>(X, T, V, S);
    sigdp_kernel<<<NSER * NSER, 32, 0, stream>>>(T, V, S, Kmat);
    solve_kernel<<<1, 256, 0, stream>>>(Kmat, Y, preds, alphas);
}